// Decoder_48627619725499
// MI455X (gfx1250) — compile-verified
//
#include <hip/hip_runtime.h>

typedef float v2f __attribute__((ext_vector_type(2)));
typedef float v8f __attribute__((ext_vector_type(8)));

// Problem constants (from reference)
constexpr int BINS    = 257;
constexpr int FRAME   = 512;
constexpr int HOP     = 128;
constexpr int NBATCH  = 16;
constexpr int NFRAMES = 4000;
constexpr int ROWS    = NBATCH * NFRAMES;             // 64000 GEMM rows
constexpr int OUTL    = (NFRAMES - 1) * HOP + FRAME - 1; // 512383 (last sample dropped)

// Tiling
constexpr int MTILE = 128;   // frames per workgroup
constexpr int NTILE = 64;    // samples per workgroup
constexpr int KP    = 260;   // padded K stride in LDS (bank-conflict-free b64 reads)

constexpr float TWO_PI_OVER_N = 6.28318530717958647692f / 512.0f;

__device__ __forceinline__ float hann512(int t) {
    // periodic Hann: 0.5 - 0.5*cos(2*pi*t/512)
    return 0.5f - 0.5f * __cosf((float)t * TWO_PI_OVER_N);
}

__global__ void zero_kernel(float* __restrict__ out, int n) {
    int i = blockIdx.x * blockDim.x + threadIdx.x;
    int stride = gridDim.x * blockDim.x;
    for (; i < n; i += stride) out[i] = 0.0f;
}

__global__ __launch_bounds__(256)
void istft_wmma_kernel(const float* __restrict__ re,
                       const float* __restrict__ im,
                       float* __restrict__ out) {
    // B-tile in LDS, stored transposed: Br[n_local][k], k-stride padded to 260.
    __shared__ float Br[NTILE][KP];
    __shared__ float Bi[NTILE][KP];
    __shared__ float wf[NTILE];   // win[n]/denom[n%128]/512, folded scale

    const int tid  = threadIdx.x;
    const int nb0  = blockIdx.y * NTILE;

    // --- 1) inverse-window factor for our 64 sample columns ---
    if (tid < NTILE) {
        const int n = nb0 + tid;
        const float win = hann512(n);
        const int j = n & (HOP - 1);
        float denom = 0.0f;
        #pragma unroll
        for (int o = 0; o < 4; ++o) {
            const float w = hann512(j + o * HOP);
            denom += w * w;
        }
        wf[tid] = win / denom * (1.0f / 512.0f);
    }
    __syncthreads();

    // --- 2) synthesize folded DFT basis tile: k = 0..255 (k=256 handled in epilogue) ---
    for (int idx = tid; idx < NTILE * 256; idx += 256) {
        const int k  = idx >> 6;       // 0..255
        const int nl = idx & (NTILE - 1);
        const int n  = nb0 + nl;
        const int p  = (k * n) & 511;  // phase index mod 512 (exact in int)
        float s, c;
        __sincosf((float)p * TWO_PI_OVER_N, &s, &c);
        const float sk = (k == 0) ? 1.0f : 2.0f;
        const float w  = wf[nl];
        Br[nl][k] =  sk * c * w;
        Bi[nl][k] = -sk * s * w;
    }
    __syncthreads();

    // --- 3) WMMA GEMM: each wave computes a 16(frames) x 64(samples) tile ---
    const int l      = tid & 31;
    const int wv     = tid >> 5;
    const int half   = l >> 4;        // 0: lanes 0-15, 1: lanes 16-31
    const int lane16 = l & 15;
    const int m0     = blockIdx.x * MTILE + wv * 16;

    const float* pr = re + (size_t)(m0 + lane16) * BINS + half * 2;
    const float* pi = im + (size_t)(m0 + lane16) * BINS + half * 2;

    v8f acc[4];
    #pragma unroll
    for (int st = 0; st < 4; ++st)
        acc[st] = (v8f){0.f, 0.f, 0.f, 0.f, 0.f, 0.f, 0.f, 0.f};

    #pragma unroll 2
    for (int k0 = 0; k0 < 256; k0 += 4) {
        // A fragments (rows have odd 257-float stride -> 4B-aligned b32 loads)
        v2f ar, ai;
        ar.x = pr[k0];  ar.y = pr[k0 + 1];
        ai.x = pi[k0];  ai.y = pi[k0 + 1];

        const int kb = k0 + half * 2;
        #pragma unroll
        for (int st = 0; st < 4; ++st) {
            const int nl = st * 16 + lane16;
            const v2f br = *(const v2f*)(&Br[nl][kb]);   // 8B-aligned ds_load
            const v2f bi = *(const v2f*)(&Bi[nl][kb]);
            acc[st] = __builtin_amdgcn_wmma_f32_16x16x4_f32(
                false, ar, false, br, (short)0, acc[st], false, false);
            acc[st] = __builtin_amdgcn_wmma_f32_16x16x4_f32(
                false, ai, false, bi, (short)0, acc[st], false, false);
        }
    }

    // --- 4) epilogue: Nyquist bin k=256 (coef = (-1)^n * w[n]/512, imag part is zero) ---
    float r256[8];
    #pragma unroll
    for (int v = 0; v < 8; ++v)
        r256[v] = re[(size_t)(m0 + v + 8 * half) * BINS + 256];

    #pragma unroll
    for (int st = 0; st < 4; ++st) {
        const int nl = st * 16 + lane16;
        const int n  = nb0 + nl;
        const float lc = ((n & 1) ? -1.0f : 1.0f) * wf[nl];
        #pragma unroll
        for (int v = 0; v < 8; ++v)
            acc[st][v] += r256[v] * lc;
    }

    // --- 5) overlap-add: atomic scatter into [B, OUTL] output ---
    #pragma unroll
    for (int st = 0; st < 4; ++st) {
        const int n = nb0 + st * 16 + lane16;
        #pragma unroll
        for (int v = 0; v < 8; ++v) {
            const int rowm = m0 + v + 8 * half;
            const int b = rowm / NFRAMES;
            const int f = rowm - b * NFRAMES;
            const int pos = f * HOP + n;
            if (pos < OUTL)  // drops only (f=3999, n=511) per batch, matching [:, :-1]
                atomicAdd(out + (size_t)b * OUTL + pos, acc[st][v]);
        }
    }
}

extern "C" void kernel_launch(void* const* d_in, const int* in_sizes, int n_in,
                              void* d_out, int out_size, void* d_ws, size_t ws_size,
                              hipStream_t stream) {
    (void)in_sizes; (void)n_in; (void)d_ws; (void)ws_size;
    const float* re = (const float*)d_in[0];
    const float* im = (const float*)d_in[1];
    float* out = (float*)d_out;

    // Zero output (it is overlap-add accumulated with atomics).
    zero_kernel<<<2048, 256, 0, stream>>>(out, out_size);

    dim3 grid(ROWS / MTILE, FRAME / NTILE);   // 500 x 8
    istft_wmma_kernel<<<grid, 256, 0, stream>>>(re, im, out);
}